// SSM_15547781611899
// MI455X (gfx1250) — compile-verified
//
#include <hip/hip_runtime.h>

// ---------------------------------------------------------------------------
// FFT convolution (Hyena/SSM long conv) for MI455X / gfx1250.
//   y[b,:,d] = irfft( rfft(u[b,:,d], 2L) * rfft(k[d,:], 2L) )[:L] + u[b,:,d]*skip[d]
// One workgroup per (b,d). 16384-point complex FFT fully in LDS.
// - Radix-16 stages = 16x16 complex matmuls on V_WMMA_F32_16X16X4_F32.
// - Forward DIF / inverse DIT => no digit-reversal pass; pointwise product in
//   the digit-reversed domain.
// - First forward stage exploits real input + zero padding (4 WMMA/batch).
// - Last inverse stage computes real part of the first half only (8 WMMA/batch).
// - Strided u gather / y scatter + contiguous k row are moved by the Tensor
//   Data Mover (tensor_load_to_lds / tensor_store_from_lds); TDM pad_enable
//   reproduces the LDS bank-padding layout in hardware.
// ---------------------------------------------------------------------------

typedef float v2f __attribute__((ext_vector_type(2)));
typedef float v8f __attribute__((ext_vector_type(8)));
typedef unsigned int u32x4 __attribute__((ext_vector_type(4)));
typedef unsigned int u32x8 __attribute__((ext_vector_type(8)));

#define NFFT  16384
#define LSEQ  8192
#define DCH   1024
#define BB    4
#define PADN  (NFFT + NFFT / 64)          // 16640 floats per array
#define PD(i) ((i) + ((i) >> 6))          // 1 dword pad per 64 (matches TDM pad cfg)

#define TWO_PI_F 6.28318530717958647692f

// LDS byte offsets of the four arrays (dynamic LDS base is offset 0: no static LDS).
#define UR_OFF 0u
#define UI_OFF (PADN * 4u)
#define KR_OFF (2u * PADN * 4u)
#define KI_OFF (3u * PADN * 4u)

__device__ __forceinline__ v8f wmma4(v2f a, v2f b, v8f c) {
  // D = A(16x4, f32) * B(4x16, f32) + C(16x16, f32)
  return __builtin_amdgcn_wmma_f32_16x16x4_f32(false, a, false, b, (short)0, c,
                                               false, false);
}

// ---------------- Tensor Data Mover helpers (CDNA5 TDM, D# per ISA 08 §8) ----
__device__ __forceinline__ u32x4 tdm_g0(unsigned lds_addr,
                                        unsigned long long gaddr) {
  u32x4 g;
  g[0] = 1u;                                    // count=1 valid, user mode
  g[1] = lds_addr;                              // LDS byte address
  g[2] = (unsigned)(gaddr & 0xFFFFFFFFu);       // global_addr[31:0]
  g[3] = (unsigned)((gaddr >> 32) & 0x01FFFFFFu) | (2u << 30);  // [56:32], type=2
  return g;
}

// 2D descriptor group1: data_size=4B, optional LDS padding (1 dword per 64).
__device__ __forceinline__ u32x8 tdm_g1(unsigned pad_en, unsigned tensor_d0,
                                        unsigned tensor_d1, unsigned tile_d0,
                                        unsigned tile_d1,
                                        unsigned long long stride0) {
  u32x8 g;
  g[0] = (2u << 16) | (pad_en << 20) | ((pad_en ? 5u : 0u) << 22); // pad: 64dw interval, 1dw amount
  g[1] = (tensor_d0 & 0xFFFFu) << 16;
  g[2] = ((tensor_d0 >> 16) & 0xFFFFu) | ((tensor_d1 & 0xFFFFu) << 16);
  g[3] = ((tensor_d1 >> 16) & 0xFFFFu) | ((tile_d0 & 0xFFFFu) << 16);
  g[4] = (tile_d1 & 0xFFFFu);                   // tile_dim2 = 0 (2D)
  g[5] = (unsigned)(stride0 & 0xFFFFFFFFu);     // tensor_dim0_stride[31:0]
  g[6] = (unsigned)((stride0 >> 32) & 0xFFFFu); // stride0[47:32]; dim1_stride=0
  g[7] = 0u;
  return g;
}

__device__ __forceinline__ void tdm_load(u32x4 g0, u32x8 g1) {
  asm volatile("tensor_load_to_lds %0, %1" ::"s"(g0), "s"(g1) : "memory");
}
__device__ __forceinline__ void tdm_store(u32x4 g0, u32x8 g1) {
  asm volatile("tensor_store_from_lds %0, %1" ::"s"(g0), "s"(g1) : "memory");
}

// ---------------- F16 DFT matrix fragments (A-operand layout) ---------------
// lane L holds F16[m][kk], m = L&15, kk = 4c + 2*(L>>4) + v.
__device__ __forceinline__ void make_f16(int lane, float sgn, v2f fr[4],
                                         v2f fi[4], v2f fin[4]) {
  const float w = sgn * (TWO_PI_F / 16.0f);
  const int m  = lane & 15;
  const int hk = (lane >> 4) << 1;
#pragma unroll
  for (int c = 0; c < 4; ++c) {
#pragma unroll
    for (int v = 0; v < 2; ++v) {
      int kk = 4 * c + hk + v;
      float s, co;
      __sincosf(w * (float)(m * kk), &s, &co);
      fr[c][v]  = co;
      fi[c][v]  = s;
      fin[c][v] = -s;  // pre-negated (f32 WMMA has no A/B negate modifier)
    }
  }
}

// Generic in-place radix-16 batch (16 butterflies) at stride S, block 16S.
// twIn=false: DIF fwd, twiddle on output. twIn=true: DIT inv, twiddle on input.
__device__ __forceinline__ void r16_batch(float* Xr, float* Xi, int g, int S,
                                          int sh, int lane, const v2f fr[4],
                                          const v2f fi[4], const v2f fin[4],
                                          float twAng, bool twIn) {
  const int n    = lane & 15;
  const int half = lane >> 4;
  const int bf   = (g << 4) + n;
  const int j    = bf & (S - 1);
  const int base = ((bf >> sh) << (sh + 4)) + j;

  v8f cr = {}, ci = {};
#pragma unroll
  for (int c = 0; c < 4; ++c) {
    v2f br, bi;
#pragma unroll
    for (int v = 0; v < 2; ++v) {
      int t = 4 * c + 2 * half + v;
      int a = PD(base + t * S);
      float xr = Xr[a], xm = Xi[a];
      if (twIn) {
        float s, co;
        __sincosf(twAng * (float)(j * t), &s, &co);
        float r = xr * co - xm * s;
        xm      = xr * s + xm * co;
        xr      = r;
      }
      br[v] = xr;
      bi[v] = xm;
    }
    cr = wmma4(fr[c],  br, cr);
    cr = wmma4(fin[c], bi, cr);
    ci = wmma4(fr[c],  bi, ci);
    ci = wmma4(fi[c],  br, ci);
  }
#pragma unroll
  for (int v = 0; v < 8; ++v) {
    int t = v + 8 * half;
    float yr = cr[v], yi = ci[v];
    if (!twIn) {
      float s, co;
      __sincosf(twAng * (float)(j * t), &s, &co);
      float r = yr * co - yi * s;
      yi      = yr * s + yi * co;
      yr      = r;
    }
    int a = PD(base + t * S);
    Xr[a] = yr;
    Xi[a] = yi;
  }
}

// First forward stage (S=1024): input is real and zero above L -> only rows
// t=0..7 of Xr are read; 4 WMMAs per batch. Writes full complex output.
__device__ __forceinline__ void r16_first(float* Xr, float* Xi, int g, int lane,
                                          const v2f fr[4], const v2f fi[4]) {
  const int n    = lane & 15;
  const int half = lane >> 4;
  const int base = (g << 4) + n;  // S=1024: j=bf, single block
  v8f cr = {}, ci = {};
#pragma unroll
  for (int c = 0; c < 2; ++c) {
    v2f br;
#pragma unroll
    for (int v = 0; v < 2; ++v) {
      int t = 4 * c + 2 * half + v;  // 0..7 (idx < L only)
      br[v] = Xr[PD(base + (t << 10))];
    }
    cr = wmma4(fr[c], br, cr);
    ci = wmma4(fi[c], br, ci);
  }
  const float twA = -TWO_PI_F / (float)NFFT;
#pragma unroll
  for (int v = 0; v < 8; ++v) {
    int t = v + 8 * half;
    float s, co;
    __sincosf(twA * (float)(base * t), &s, &co);
    int a = PD(base + (t << 10));
    Xr[a] = cr[v] * co - ci[v] * s;
    Xi[a] = cr[v] * s + ci[v] * co;
  }
}

// Last inverse stage (S=1024): only the real part of idx<L is needed ->
// single accumulator (8 WMMAs), stores rows t=0..7 real only.
__device__ __forceinline__ void r16_last(float* Xr, float* Xi, int g, int lane,
                                         const v2f fr[4], const v2f fi[4],
                                         const v2f fin[4]) {
  const int n    = lane & 15;
  const int half = lane >> 4;
  const int base = (g << 4) + n;  // j = bf
  const float twA = +TWO_PI_F / (float)NFFT;
  v8f cr = {};
#pragma unroll
  for (int c = 0; c < 4; ++c) {
    v2f br, bi;
#pragma unroll
    for (int v = 0; v < 2; ++v) {
      int t = 4 * c + 2 * half + v;
      int a = PD(base + (t << 10));
      float xr = Xr[a], xm = Xi[a];
      float s, co;
      __sincosf(twA * (float)(base * t), &s, &co);
      br[v] = xr * co - xm * s;
      bi[v] = xr * s + xm * co;
    }
    cr = wmma4(fr[c],  br, cr);
    cr = wmma4(fin[c], bi, cr);
  }
  if (half == 0) {  // rows t=0..7 -> idx = base + t*1024 < 8192
#pragma unroll
    for (int v = 0; v < 8; ++v) Xr[PD(base + (v << 10))] = cr[v];
  }
}

// Radix-4 stage, stride 1, no twiddles. isgn=-1 fwd, +1 inv.
__device__ __forceinline__ void radix4_stage(float* Xr, float* Xi, int tid,
                                             float isgn) {
  for (int q = tid; q < NFFT / 4; q += 256) {
    int b0 = q << 2;
    int a0 = PD(b0), a1 = PD(b0 + 1), a2 = PD(b0 + 2), a3 = PD(b0 + 3);
    float x0r = Xr[a0], x0i = Xi[a0];
    float x1r = Xr[a1], x1i = Xi[a1];
    float x2r = Xr[a2], x2i = Xi[a2];
    float x3r = Xr[a3], x3i = Xi[a3];
    float e0r = x0r + x2r, e0i = x0i + x2i;
    float e1r = x0r - x2r, e1i = x0i - x2i;
    float o0r = x1r + x3r, o0i = x1i + x3i;
    float o1r = x1r - x3r, o1i = x1i - x3i;
    Xr[a0] = e0r + o0r;        Xi[a0] = e0i + o0i;
    Xr[a2] = e0r - o0r;        Xi[a2] = e0i - o0i;
    Xr[a1] = e1r - isgn * o1i; Xi[a1] = e1i + isgn * o1r;
    Xr[a3] = e1r + isgn * o1i; Xi[a3] = e1i - isgn * o1r;
  }
}

__global__ __launch_bounds__(256) void ssm_fftconv_kernel(
    const float* __restrict__ u, const float* __restrict__ kfil,
    const float* __restrict__ skip, float* __restrict__ out) {
  extern __shared__ float smem[];
  float* Ur = smem;
  float* Ui = Ur + PADN;
  float* Kr = Ui + PADN;
  float* Ki = Kr + PADN;

  const int wg   = blockIdx.x;
  const int d    = wg & (DCH - 1);
  const int b    = wg >> 10;
  const int tid  = threadIdx.x;
  const int lane = tid & 31;
  const int wave = tid >> 5;

  const float* up = u + (size_t)b * LSEQ * DCH + d;  // stride-DCH column
  const float* kp = kfil + (size_t)d * LSEQ;         // contiguous row (KD==L)

  // ---- TDM DMA: strided u column + contiguous k row -> padded LDS layout ----
  if (wave == 0) {
    // u[b,:,d]: 2D tile 1 x 8192, row stride DCH elements, pad 1dw/64dw.
    tdm_load(tdm_g0(UR_OFF, (unsigned long long)(uintptr_t)up),
             tdm_g1(1u, 1u, LSEQ, 1u, LSEQ, (unsigned long long)DCH));
    // k[d,:]: 1 row of 8192 contiguous elements, same LDS padding.
    tdm_load(tdm_g0(KR_OFF, (unsigned long long)(uintptr_t)kp),
             tdm_g1(1u, LSEQ, 1u, LSEQ, 1u, (unsigned long long)LSEQ));
    __builtin_amdgcn_s_wait_tensorcnt(0);
  }
  __syncthreads();
  // No zero-fill / imag-fill needed: the first stage reads only Xr[idx<L] and
  // writes every element of both Xr and Xi.

  v2f fr[4], fi[4], fin[4];
  make_f16(lane, -1.0f, fr, fi, fin);

  // ---- forward DIF: special S=1024 (real, half-K), then S=64, 4, radix-4 ----
  for (int i2 = 0; i2 < 8; ++i2) {
    int g = wave * 8 + i2;  // 64 batches of 16 butterflies over 8 waves
    r16_first(Ur, Ui, g, lane, fr, fi);
    r16_first(Kr, Ki, g, lane, fr, fi);
  }
  __syncthreads();
  {
    const int Ss[2] = {64, 4};
    const int sh[2] = {6, 2};
#pragma unroll
    for (int s = 0; s < 2; ++s) {
      const float twA = -TWO_PI_F / (float)(16 * Ss[s]);
      for (int i2 = 0; i2 < 8; ++i2) {
        int g = wave * 8 + i2;
        r16_batch(Ur, Ui, g, Ss[s], sh[s], lane, fr, fi, fin, twA, false);
        r16_batch(Kr, Ki, g, Ss[s], sh[s], lane, fr, fi, fin, twA, false);
      }
      __syncthreads();
    }
  }
  radix4_stage(Ur, Ui, tid, -1.0f);
  radix4_stage(Kr, Ki, tid, -1.0f);
  __syncthreads();

  // ---- pointwise complex product (digit-reversed frequency domain) ----
  for (int i = tid; i < NFFT; i += 256) {
    int a = PD(i);
    float ur = Ur[a], ui = Ui[a], kr = Kr[a], ki = Ki[a];
    Ur[a] = ur * kr - ui * ki;
    Ui[a] = ur * ki + ui * kr;
  }
  __syncthreads();

  // ---- inverse DIT: radix-4, S=4, S=64, then special S=1024 (real out) ----
  make_f16(lane, +1.0f, fr, fi, fin);
  radix4_stage(Ur, Ui, tid, +1.0f);
  __syncthreads();
  {
    const int Ss[2] = {4, 64};
    const int sh[2] = {2, 6};
#pragma unroll
    for (int s = 0; s < 2; ++s) {
      const float twA = +TWO_PI_F / (float)(16 * Ss[s]);
      for (int i2 = 0; i2 < 8; ++i2) {
        int g = wave * 8 + i2;
        r16_batch(Ur, Ui, g, Ss[s], sh[s], lane, fr, fi, fin, twA, true);
      }
      __syncthreads();
    }
  }
  for (int i2 = 0; i2 < 8; ++i2) {
    int g = wave * 8 + i2;
    r16_last(Ur, Ui, g, lane, fr, fi, fin);
  }
  __syncthreads();

  // ---- scale 1/N + skip, compact into (dead) Kr linearly, TDM scatter out ----
  const float skipv = skip[d];
  const float invN  = 1.0f / (float)NFFT;
  for (int i = tid; i < LSEQ; i += 256) {
    Kr[i] = Ur[PD(i)] * invN + up[(size_t)i * DCH] * skipv;  // linear, no pad
  }
  __syncthreads();
  if (wave == 0) {
    const float* op = out + (size_t)b * LSEQ * DCH + d;
    tdm_store(tdm_g0(KR_OFF, (unsigned long long)(uintptr_t)op),
              tdm_g1(0u, 1u, LSEQ, 1u, LSEQ, (unsigned long long)DCH));
    __builtin_amdgcn_s_wait_tensorcnt(0);  // keep LDS alive until DMA done
  }
}

extern "C" void kernel_launch(void* const* d_in, const int* in_sizes, int n_in,
                              void* d_out, int out_size, void* d_ws,
                              size_t ws_size, hipStream_t stream) {
  (void)in_sizes; (void)n_in; (void)out_size; (void)d_ws; (void)ws_size;
  const float* u    = (const float*)d_in[0];
  const float* kfil = (const float*)d_in[1];
  const float* skip = (const float*)d_in[2];
  float* out        = (float*)d_out;

  const size_t smem_bytes = (size_t)4 * PADN * sizeof(float);  // 266,240 B < 320 KB
  static_assert((size_t)4 * PADN * sizeof(float) <= (size_t)320 * 1024,
                "LDS budget exceeded");
  hipFuncSetAttribute((const void*)ssm_fftconv_kernel,
                      hipFuncAttributeMaxDynamicSharedMemorySize,
                      (int)smem_bytes);

  ssm_fftconv_kernel<<<dim3(BB * DCH), dim3(256), smem_bytes, stream>>>(
      u, kfil, skip, out);
}